// ConvModel_35983236006617
// MI455X (gfx1250) — compile-verified
//
#include <hip/hip_runtime.h>
#include <hip/hip_bf16.h>

#define IN_C   64
#define OUT_C  128
#define IMG_H  224
#define IMG_W  224
#define KDIM   576            // IN_C * 9
#define K_HALF 288            // 32 channels * 9 per pass
#define HC     32             // channels per pass
#define TILE_W 32             // output pixels per workgroup
#define XS_COLS 40            // aligned staging window: cols w0-4 .. w0+35
#define XS_I_STRIDE (3 * XS_COLS)         // 120 floats per channel
#define XS_ELEMS (HC * 3 * XS_COLS)       // 3840 floats = 15360 B
#define KPAD   290            // im2col row stride; gcd(34,64)=2 -> conflict-free b64

typedef __attribute__((ext_vector_type(2))) float v2f;
typedef __attribute__((ext_vector_type(4))) float v4f;
typedef __attribute__((ext_vector_type(8))) float v8f;
typedef __attribute__((ext_vector_type(4))) int   v4i;

typedef v4i __attribute__((address_space(1)))* gptr4i;   // global int4*
typedef v4i __attribute__((address_space(3)))* lptr4i;   // LDS int4*

#if __has_builtin(__builtin_amdgcn_global_load_async_to_lds_b128)
#define HAVE_ASYNC_LDS 1
#else
#define HAVE_ASYNC_LDS 0
#endif

__global__ __launch_bounds__(256) void conv3x3_wmma_f32(
    const float* __restrict__ x,      // [32][64][224][224]
    const float* __restrict__ wgt,    // [128][64][3][3]  (flat: o*576 + k)
    const float* __restrict__ bias,   // [128]
    float* __restrict__ out)          // [32][128][224][224]
{
    __shared__ __align__(16) float xs[XS_ELEMS];        // raw halo tile (half channels)
    __shared__ __align__(16) float bm[TILE_W * KPAD];   // linear im2col panel

    const int tid  = threadIdx.x;
    const int lane = tid & 31;
    const int wave = tid >> 5;

    const int w0 = blockIdx.x * TILE_W;
    const int h  = blockIdx.y;
    const int n  = blockIdx.z;

    const int o0   = wave * 16;
    const int lm   = lane & 15;          // M index (A) / pixel column (B)
    const int koff = (lane >> 4) * 2;    // hi half-wave covers K+2,K+3

    v8f acc0 = {};   // pixels w0 + [0,16)
    v8f acc1 = {};   // pixels w0 + [16,32)

    const float* xn = x + (size_t)n * IN_C * IMG_H * IMG_W;

    // Uniform per block: staging window [h-1,h+1] x [w0-4, w0+35] fully in-bounds?
    const bool interior = (h >= 1) && (h <= IMG_H - 2) &&
                          (w0 >= TILE_W) && (w0 <= IMG_W - 2 * TILE_W);

    for (int half = 0; half < 2; ++half) {
        const int ibase = half * HC;

        // ---- stage 1: raw halo tile for channels [ibase, ibase+32) ----
        if (interior) {
            // 32 ch x 3 rows x 10 aligned float4 chunks = 960 async B128 moves
            for (int e = tid; e < HC * 30; e += 256) {
                int i   = e / 30;
                int rem = e - i * 30;
                int r   = rem / 10;
                int q   = rem - r * 10;
                const float* gp = xn + (size_t)(ibase + i) * (IMG_H * IMG_W)
                                + (h + r - 1) * IMG_W + (w0 - 4) + q * 4;
                float* lp = &xs[i * XS_I_STRIDE + r * XS_COLS + q * 4];
#if HAVE_ASYNC_LDS
                __builtin_amdgcn_global_load_async_to_lds_b128(
                    (gptr4i)gp, (lptr4i)lp, 0, 0);
#else
                *(v4f*)lp = *(const v4f*)gp;
#endif
            }
#if HAVE_ASYNC_LDS
#if __has_builtin(__builtin_amdgcn_s_wait_asynccnt)
            __builtin_amdgcn_s_wait_asynccnt(0);
#else
            asm volatile("s_wait_asynccnt 0x0" ::: "memory");
#endif
#endif
        } else {
            // boundary blocks: scalar zero-padded fill, same xs layout
            for (int t = tid; t < XS_ELEMS; t += 256) {
                int i   = t / XS_I_STRIDE;
                int rem = t - i * XS_I_STRIDE;
                int r   = rem / XS_COLS;
                int c   = rem - r * XS_COLS;
                int row = h + r - 1;
                int col = w0 - 4 + c;
                float v = 0.0f;
                if ((unsigned)row < IMG_H && (unsigned)col < IMG_W)
                    v = xn[(size_t)(ibase + i) * (IMG_H * IMG_W) + row * IMG_W + col];
                xs[t] = v;
            }
        }
        __syncthreads();   // also guarantees prior half's bm readers are done

        // ---- stage 2: expand to linear im2col panel bm[col][k] ----
        {
            const int colpix = tid & 31;  // pixel column 0..31
            const int ib     = tid >> 5;  // 0..7
#pragma unroll
            for (int m = 0; m < 4; ++m) {
                const int i = ib + 8 * m;             // channel-in-half 0..31
                // global col (w0+colpix+c-1) lives at xs col (colpix+c+3)
                const float* src = &xs[i * XS_I_STRIDE + 3 + colpix];
                float*       dst = &bm[colpix * KPAD + i * 9];
#pragma unroll
                for (int r = 0; r < 3; ++r)
#pragma unroll
                    for (int c = 0; c < 3; ++c)
                        dst[r * 3 + c] = src[r * XS_COLS + c];
            }
        }
        __syncthreads();

        // ---- hot loop: 72 k-steps, all-linear addressing ----
        const float* wptr = wgt + (size_t)(o0 + lm) * KDIM + half * K_HALF + koff;
        const float* b0p  = &bm[lm * KPAD + koff];
        const float* b1p  = &bm[(lm + 16) * KPAD + koff];

        __builtin_prefetch(wptr, 0, 3);              // global_prefetch_b8
        __builtin_prefetch(wptr + 160, 0, 3);

#pragma unroll 4
        for (int k0 = 0; k0 < K_HALF; k0 += 4) {
            v2f a  = *(const v2f*)(wptr + k0);       // A: (M=lm, K={k,k+1}) per half-wave
            v2f b0 = *(const v2f*)(b0p + k0);        // ds_load_b64, conflict-free
            v2f b1 = *(const v2f*)(b1p + k0);
            acc0 = __builtin_amdgcn_wmma_f32_16x16x4_f32(
                       false, a, false, b0, (short)0, acc0, false, false);
            acc1 = __builtin_amdgcn_wmma_f32_16x16x4_f32(
                       false, a, false, b1, (short)0, acc1, false, false);
        }
        __syncthreads();   // protect xs/bm before next half's refill
    }

    // ---- epilogue: C/D layout VGPR j: lanes0-15 -> M=j, lanes16-31 -> M=j+8 ----
    const int mhi = (lane >> 4) * 8;
    float* obase = out + (size_t)n * OUT_C * IMG_H * IMG_W + (size_t)h * IMG_W;
#pragma unroll
    for (int j = 0; j < 8; ++j) {
        int o = o0 + j + mhi;
        float bv = bias[o];
        __builtin_nontemporal_store(acc0[j] + bv,
            &obase[(size_t)o * (IMG_H * IMG_W) + w0 + lm]);
        __builtin_nontemporal_store(acc1[j] + bv,
            &obase[(size_t)o * (IMG_H * IMG_W) + w0 + 16 + lm]);
    }
}

extern "C" void kernel_launch(void* const* d_in, const int* in_sizes, int n_in,
                              void* d_out, int out_size, void* d_ws, size_t ws_size,
                              hipStream_t stream) {
    (void)in_sizes; (void)n_in; (void)out_size; (void)d_ws; (void)ws_size;
    const float* x    = (const float*)d_in[0];
    const float* wgt  = (const float*)d_in[1];
    const float* bias = (const float*)d_in[2];
    float* out = (float*)d_out;

    dim3 grid(IMG_W / TILE_W, IMG_H, 32);   // (7, 224, 32)
    dim3 block(256);                        // 8 wave32 waves
    conv3x3_wmma_f32<<<grid, block, 0, stream>>>(x, wgt, bias, out);
}